// MMD_Loss_24644522344587
// MI455X (gfx1250) — compile-verified
//
#include <hip/hip_runtime.h>
#include <stdint.h>

typedef __attribute__((ext_vector_type(2))) float v2f;
typedef __attribute__((ext_vector_type(8))) float v8f;

#define NROWS   8192
#define HALF_N  4096
#define DIMK    256
#define BM      128
#define BN      128
#define KBLK    16
#define NKB     (DIMK / KBLK)   // 16 K-blocks
#define LDSS    20              // padded LDS row stride in floats (bank-conflict-free, 16B aligned)
#define TILEF   (BM * LDSS)     // floats per LDS tile buffer

struct Ws {
    double sumL2;        // offset 0
    double quad[4];      // offset 8  (xx, xy, yx, yy by row/col quadrant bits)
    int    maxBits;      // offset 40 (float-as-int max, valid: L2 >= 0)
    float  negInvBw[5];  // offset 44
    float  sq[NROWS];    // offset 64
};

// ---------------------------------------------------------------------------
// Kernel 1: row squared-norms + accumulator init (block 0)
// ---------------------------------------------------------------------------
__global__ __launch_bounds__(256) void mmd_sq_init_kernel(
    const float* __restrict__ src, const float* __restrict__ tgt, Ws* __restrict__ ws) {
    if (blockIdx.x == 0 && threadIdx.x < 8) {
        if (threadIdx.x == 0) { ws->sumL2 = 0.0; ws->maxBits = 0; }
        if (threadIdx.x < 4)  { ws->quad[threadIdx.x] = 0.0; }
    }
    const int row  = blockIdx.x * 8 + (threadIdx.x >> 5);
    const int lane = threadIdx.x & 31;
    const float* p = (row < HALF_N) ? (src + (size_t)row * DIMK)
                                    : (tgt + (size_t)(row - HALF_N) * DIMK);
    float s = 0.0f;
#pragma unroll
    for (int it = 0; it < 2; ++it) {
        float4 v = *(const float4*)(p + it * 128 + lane * 4);
        s += v.x * v.x + v.y * v.y + v.z * v.z + v.w * v.w;
    }
#pragma unroll
    for (int off = 16; off > 0; off >>= 1) s += __shfl_down(s, off, 32);
    if (lane == 0) ws->sq[row] = s;
}

// ---------------------------------------------------------------------------
// Async global -> LDS copy (CDNA5 ASYNCcnt path, bypasses VGPRs).
// ldsOff = workgroup-relative LDS byte offset (low 32 bits of flat shared ptr).
// ---------------------------------------------------------------------------
__device__ __forceinline__ void async_copy_b128(unsigned ldsOff, const float* gptr) {
    uint64_t ga = (uint64_t)(uintptr_t)gptr;
    asm volatile("global_load_async_to_lds_b128 %0, %1, off"
                 :: "v"(ldsOff), "v"(ga) : "memory");
}
__device__ __forceinline__ void wait_async_le4() {
    asm volatile("s_wait_asynccnt 0x4" ::: "memory");
}
__device__ __forceinline__ void wait_async_le0() {
    asm volatile("s_wait_asynccnt 0x0" ::: "memory");
}

// ---------------------------------------------------------------------------
// Fused tiled GEMM (T * T^T) + L2 epilogue, double-buffered async LDS staging.
// PASS==1: global sum + max of L2.  PASS==2: quadrant sums of 5-kernel exp.
// ---------------------------------------------------------------------------
template <int PASS>
__global__ __launch_bounds__(256) void mmd_gemm_kernel(
    const float* __restrict__ src, const float* __restrict__ tgt, Ws* __restrict__ ws) {
    __shared__ float lA[2 * TILEF];
    __shared__ float lB[2 * TILEF];
    __shared__ float redS[8];
    __shared__ float redM[8];

    const int t     = threadIdx.x;
    const int wave  = t >> 5;
    const int lane  = t & 31;
    const int laneN = lane & 15;
    const int kb    = (lane >> 4) << 1;  // K sub-offset per lane half: 0 or 2
    const int mbase = (lane >> 4) << 3;  // C/D layout: M base 0 or 8

    const int rowBase = blockIdx.y * BM;
    const int colBase = blockIdx.x * BN;

    // 4096 % 128 == 0 -> a tile never straddles the source/target boundary.
    const float* Abase = (rowBase < HALF_N) ? (src + (size_t)rowBase * DIMK)
                                            : (tgt + (size_t)(rowBase - HALF_N) * DIMK);
    const float* Bbase = (colBase < HALF_N) ? (src + (size_t)colBase * DIMK)
                                            : (tgt + (size_t)(colBase - HALF_N) * DIMK);

    v8f acc[8];
#pragma unroll
    for (int n = 0; n < 8; ++n) { v8f z = {}; acc[n] = z; }

    const int lr = t >> 2;        // 0..63 : row within tile (x2 halves)
    const int lc = (t & 3) << 2;  // 0,4,8,12 : float4 column within K-block

    // Per-thread async staging: 2x A-halves + 2x B-halves = 4 B128 ops / buffer.
    const unsigned ldsA0 = (unsigned)(uintptr_t)&lA[lr * LDSS + lc];
    const unsigned ldsA1 = (unsigned)(uintptr_t)&lA[(lr + 64) * LDSS + lc];
    const unsigned ldsB0 = (unsigned)(uintptr_t)&lB[lr * LDSS + lc];
    const unsigned ldsB1 = (unsigned)(uintptr_t)&lB[(lr + 64) * LDSS + lc];
    const float* gA0 = Abase + (size_t)lr * DIMK + lc;
    const float* gA1 = Abase + (size_t)(lr + 64) * DIMK + lc;
    const float* gB0 = Bbase + (size_t)lr * DIMK + lc;
    const float* gB1 = Bbase + (size_t)(lr + 64) * DIMK + lc;

    // Prologue: stage K-block 0 into buffer 0.
    async_copy_b128(ldsA0, gA0);
    async_copy_b128(ldsA1, gA1);
    async_copy_b128(ldsB0, gB0);
    async_copy_b128(ldsB1, gB1);

    for (int i = 0; i < NKB; ++i) {
        const int bufOff = (i & 1) * TILEF;
        // Issue next K-block into the other buffer, then wait for current.
        if (i + 1 < NKB) {
            const unsigned nOff = (unsigned)(((i + 1) & 1) * TILEF * sizeof(float));
            const int nk = (i + 1) * KBLK;
            async_copy_b128(ldsA0 + nOff, gA0 + nk);
            async_copy_b128(ldsA1 + nOff, gA1 + nk);
            async_copy_b128(ldsB0 + nOff, gB0 + nk);
            async_copy_b128(ldsB1 + nOff, gB1 + nk);
            wait_async_le4();  // in-order completion: current buffer has landed
        } else {
            wait_async_le0();
        }
        __syncthreads();  // all waves' staging for this buffer is visible

        const float* aRow = &lA[bufOff + (wave * 16 + laneN) * LDSS + kb];
        const float* bCol = &lB[bufOff + laneN * LDSS + kb];
#pragma unroll
        for (int kk = 0; kk < KBLK; kk += 4) {
            v2f a = *(const v2f*)(aRow + kk);
            v2f bf[8];
#pragma unroll
            for (int n = 0; n < 8; ++n)
                bf[n] = *(const v2f*)(bCol + n * 16 * LDSS + kk);
#pragma unroll
            for (int n = 0; n < 8; ++n)
                acc[n] = __builtin_amdgcn_wmma_f32_16x16x4_f32(
                    /*neg_a=*/false, a, /*neg_b=*/false, bf[n],
                    /*c_mod=*/(short)0, acc[n], /*reuse_a=*/false, /*reuse_b=*/false);
        }
        __syncthreads();  // done reading buffer before it is overwritten
    }

    // ---- epilogue: L2[i][j] = sq[i] + sq[j] - 2*dot, clamped at 0 ----
    const int gRow0 = rowBase + wave * 16 + mbase;  // rows gRow0 .. gRow0+7
    float sqr[8], sqc[8];
#pragma unroll
    for (int v = 0; v < 8; ++v) sqr[v] = ws->sq[gRow0 + v];
#pragma unroll
    for (int n = 0; n < 8; ++n) sqc[n] = ws->sq[colBase + n * 16 + laneN];

    if (PASS == 1) {
        float s = 0.0f, mx = 0.0f;
#pragma unroll
        for (int n = 0; n < 8; ++n) {
#pragma unroll
            for (int v = 0; v < 8; ++v) {
                float l2 = fmaxf(sqr[v] + sqc[n] - 2.0f * acc[n][v], 0.0f);
                s += l2;
                mx = fmaxf(mx, l2);
            }
        }
#pragma unroll
        for (int off = 16; off > 0; off >>= 1) {
            s += __shfl_down(s, off, 32);
            mx = fmaxf(mx, __shfl_down(mx, off, 32));
        }
        if (lane == 0) { redS[wave] = s; redM[wave] = mx; }
        __syncthreads();
        if (t == 0) {
            float bs = 0.0f, bm = 0.0f;
#pragma unroll
            for (int w = 0; w < 8; ++w) { bs += redS[w]; bm = fmaxf(bm, redM[w]); }
            atomicAdd(&ws->sumL2, (double)bs);
            atomicMax(&ws->maxBits, __float_as_int(bm));
        }
    } else {
        float nib[5];
#pragma unroll
        for (int i = 0; i < 5; ++i) nib[i] = ws->negInvBw[i];
        float s = 0.0f;
#pragma unroll
        for (int n = 0; n < 8; ++n) {
#pragma unroll
            for (int v = 0; v < 8; ++v) {
                float l2 = fmaxf(sqr[v] + sqc[n] - 2.0f * acc[n][v], 0.0f);
                float e = 0.0f;
#pragma unroll
                for (int i2 = 0; i2 < 5; ++i2) e += __expf(l2 * nib[i2]);
                s += e;
            }
        }
#pragma unroll
        for (int off = 16; off > 0; off >>= 1) s += __shfl_down(s, off, 32);
        if (lane == 0) redS[wave] = s;
        __syncthreads();
        if (t == 0) {
            float bs = 0.0f;
#pragma unroll
            for (int w = 0; w < 8; ++w) bs += redS[w];
            const int q = ((rowBase >= HALF_N) ? 2 : 0) | ((colBase >= HALF_N) ? 1 : 0);
            atomicAdd(&ws->quad[q], (double)bs);
        }
    }
}

// ---------------------------------------------------------------------------
// Kernel 3: bandwidth -> -1/bw_i table
// ---------------------------------------------------------------------------
__global__ void mmd_bw_kernel(Ws* __restrict__ ws) {
    if (threadIdx.x == 0) {
        const double n = (double)NROWS;
        double bw = (ws->sumL2 / (n * n - n)) / 4.0;  // KERNEL_MUL^(KERNEL_NUM//2) = 2^2
        double m = 1.0;
        for (int i = 0; i < 5; ++i) {
            ws->negInvBw[i] = (float)(-1.0 / (bw * m + 1e-9));
            m *= 2.0;
        }
    }
}

// ---------------------------------------------------------------------------
// Kernel 5: finalize 6 scalar outputs (loss, maxL2, xx, yy, xy, yx)
// ---------------------------------------------------------------------------
__global__ void mmd_finalize_kernel(const Ws* __restrict__ ws, float* __restrict__ out) {
    if (threadIdx.x == 0) {
        const double den = (double)HALF_N * (double)HALF_N;
        double xx = ws->quad[0] / den;
        double xy = ws->quad[1] / den;
        double yx = ws->quad[2] / den;
        double yy = ws->quad[3] / den;
        out[0] = (float)(xx + yy - xy - yx);
        out[1] = __int_as_float(ws->maxBits);
        out[2] = (float)xx;
        out[3] = (float)yy;
        out[4] = (float)xy;
        out[5] = (float)yx;
    }
}

// ---------------------------------------------------------------------------
extern "C" void kernel_launch(void* const* d_in, const int* in_sizes, int n_in,
                              void* d_out, int out_size, void* d_ws, size_t ws_size,
                              hipStream_t stream) {
    const float* src = (const float*)d_in[0];
    const float* tgt = (const float*)d_in[1];
    float* out = (float*)d_out;
    Ws* ws = (Ws*)d_ws;

    mmd_sq_init_kernel<<<NROWS / 8, 256, 0, stream>>>(src, tgt, ws);
    mmd_gemm_kernel<1><<<dim3(NROWS / BN, NROWS / BM), 256, 0, stream>>>(src, tgt, ws);
    mmd_bw_kernel<<<1, 32, 0, stream>>>(ws);
    mmd_gemm_kernel<2><<<dim3(NROWS / BN, NROWS / BM), 256, 0, stream>>>(src, tgt, ws);
    mmd_finalize_kernel<<<1, 32, 0, stream>>>(ws, out);
}